// PDEInspiredModel_35631048688016
// MI455X (gfx1250) — compile-verified
//
#include <hip/hip_runtime.h>
#include <hip/hip_bf16.h>

typedef __attribute__((ext_vector_type(2))) float v2f;
typedef __attribute__((ext_vector_type(4))) float v4f;   // native vector: OK for nontemporal builtin
typedef __attribute__((ext_vector_type(8))) float v8f;

// 64 rows per block, 64 threads (2 waves, wave32). B = 131072 is divisible by 64,
// so EXEC is always all-ones (required for WMMA).
#define RPB 64

__global__ __launch_bounds__(RPB) void pde_fused_kernel(
    const float* __restrict__ x,        // (B,400)
    const float* __restrict__ wh_rec,   // (1,2)
    const float* __restrict__ wx_rec,   // (2,2)
    const float* __restrict__ wx_eval,  // (2,2)
    const float* __restrict__ wh_eval,  // (2,)
    const float* __restrict__ b_eval,   // (2,)
    const float* __restrict__ conv_w,   // (3,3,4)
    const float* __restrict__ conv_b,   // (3,)
    const float* __restrict__ fc_w,     // (3,12)
    const float* __restrict__ fc_b,     // (3,)
    float* __restrict__ out)            // (B,3)
{
    __shared__ float s_h[RPB * 101];     // h[row][0..99], stride 101 (odd -> conflict free)
    __shared__ float s_feat[RPB * 77];   // pooled feat: [ch*25+q], stride 77 (odd)
    __shared__ float s_y[RPB * 13];      // tanh'd 12-vector per row, stride 13 (odd)
    __shared__ float s_logit[RPB * 4];   // 3 logits per row (padded to 4)

    const int t = threadIdx.x;
    const long long blockRow0 = (long long)blockIdx.x * RPB;

    // ---------------- Phase 1: coalesced NT stream of x -> h (4:1 mean) ----------------
    // Each h[r][k] = mean of float4 at x[row r][4k..4k+3]; the 64*100 float4s of this
    // block are contiguous in memory, so idx = t + 64*it is perfectly coalesced.
    // x is touched exactly once across the whole grid -> non-temporal loads so the
    // 210 MB stream does not thrash WGP$/L2.
    {
        const v4f* xv = (const v4f*)x;
        const long long base4 = blockRow0 * 100;
        #pragma unroll 8
        for (int it = 0; it < 100; ++it) {
            int idx = t + it * RPB;            // 0..6399
            v4f v = __builtin_nontemporal_load(&xv[base4 + idx]);
            int r = idx / 100;
            int k = idx - r * 100;
            s_h[r * 101 + k] = (v.x + v.y + v.z + v.w) * 0.25f;
        }
    }
    __syncthreads();

    // ---------------- constants (cached scalar-ish loads) ----------------
    const float a0  = wh_rec[0],  a1  = wh_rec[1];
    const float W00 = wx_rec[0],  W01 = wx_rec[1];
    const float W10 = wx_rec[2],  W11 = wx_rec[3];
    const float we00 = wx_eval[0], we01 = wx_eval[1];
    const float we10 = wx_eval[2], we11 = wx_eval[3];
    const float whe0 = wh_eval[0], whe1 = wh_eval[1];
    const float be0  = b_eval[0],  be1  = b_eval[1];

    // ---------------- Phase 2: recurrence + eval + 4:1 mean -> feat ----------------
    // x_rec[i] (i>=2) follows: out = c @ wx_rec + hseq*wh_rec, c = previous out,
    // hseq_{i-2} = h[0] if i==2 else h[i-1].  x_rec[0]=x_rec[1]=0 matches c0=0.
    {
        const float* hrow = &s_h[t * 101];
        float* frow = &s_feat[t * 77];
        const float h0 = hrow[0];
        float hprev = 0.f;
        float c0 = 0.f, c1 = 0.f;   // x_rec[i]
        float p0 = 0.f, p1 = 0.f;   // x_rec[i-1]
        for (int q = 0; q < 25; ++q) {
            float ah = 0.f, ae0 = 0.f, ae1 = 0.f;
            #pragma unroll
            for (int s = 0; s < 4; ++s) {
                const int i = q * 4 + s;
                const float hi = hrow[i];
                if (i >= 2) {
                    const float hs = (i == 2) ? h0 : hprev;  // hseq_{i-2}
                    const float n0 = fmaf(c0, W00, fmaf(c1, W10, hs * a0));
                    const float n1 = fmaf(c0, W01, fmaf(c1, W11, hs * a1));
                    p0 = c0; p1 = c1; c0 = n0; c1 = n1;
                }
                // h_term: h_shift with [2] replaced by h[0]
                const float hterm = (i == 0) ? 0.f : ((i == 2) ? h0 : hprev);
                const float xc0 = (i >= 3) ? c0 : 0.f;
                const float xc1 = (i >= 3) ? c1 : 0.f;
                const float xp0 = (i >= 4) ? p0 : 0.f;
                const float xp1 = (i >= 4) ? p1 : 0.f;
                float e0 = fmaf(xp0, we00, fmaf(xc0, we01, fmaf(hterm, whe0, be0)));
                float e1 = fmaf(xp1, we10, fmaf(xc1, we11, fmaf(hterm, whe1, be1)));
                if (i < 2) { e0 = 0.f; e1 = 0.f; }
                ah += hi; ae0 += e0; ae1 += e1;
                hprev = hi;
            }
            frow[q]      = ah  * 0.25f;  // channel 0 = h
            frow[25 + q] = ae0 * 0.25f;  // channel 1 = ev0
            frow[50 + q] = ae1 * 0.25f;  // channel 2 = ev1
        }
    }
    // feat is per-thread-private row data -> no barrier needed before conv.

    // ---------------- Phase 3: conv(k=4) + 5:1 max-pool + tanh -> y12 ----------------
    {
        const float* frow = &s_feat[t * 77];
        float cw[3][3][4];
        #pragma unroll
        for (int o = 0; o < 3; ++o)
            #pragma unroll
            for (int ic = 0; ic < 3; ++ic)
                #pragma unroll
                for (int k = 0; k < 4; ++k)
                    cw[o][ic][k] = conv_w[(o * 3 + ic) * 4 + k];
        const float cb0 = conv_b[0], cb1 = conv_b[1], cb2 = conv_b[2];
        float* yrow = &s_y[t * 13];
        #pragma unroll
        for (int o = 0; o < 3; ++o) {
            const float cb = (o == 0) ? cb0 : ((o == 1) ? cb1 : cb2);
            #pragma unroll
            for (int q = 0; q < 4; ++q) {
                float m = -3.4e38f;
                #pragma unroll
                for (int s5 = 0; s5 < 5; ++s5) {
                    const int p = q * 5 + s5;    // p in [0,19] (y[:, :, :20])
                    float acc = cb;
                    #pragma unroll
                    for (int ic = 0; ic < 3; ++ic)
                        #pragma unroll
                        for (int k = 0; k < 4; ++k)
                            acc = fmaf(frow[ic * 25 + p + k], cw[o][ic][k], acc);
                    m = fmaxf(m, acc);
                }
                yrow[o * 4 + q] = tanhf(m);      // reshape (3,4)->(12): idx = o*4+q
            }
        }
    }
    __syncthreads();

    // ---------------- Phase 4: FC 12->3 via V_WMMA_F32_16X16X4_F32 ----------------
    // logits[r][n] = sum_c y12[r][c] * fc_w[n][c].  A = y12 (16 rows x K), B = fc_w^T
    // zero-padded to 16 cols.  K=12 = 3 accumulating K=4 WMMA steps.
    // 16x4 f32 A layout: lanes 0-15 hold K=0(v0),1(v1); lanes 16-31 hold K=2(v0),3(v1).
    // B (4x16): lanes 0-15 N=lane, K=0/1; lanes 16-31 N=lane-16, K=2/3.
    {
        const int lane = t & 31;
        const int waveId = t >> 5;
        const int M = lane & 15;                 // row within group / column N for B
        const int koff = (lane >> 4) << 1;       // 0 for lanes 0-15, 2 for lanes 16-31

        // Build the B fragments once (identical for both 16-row groups).
        // Unconditional loads with clamped row index + multiplicative zero-mask:
        // avoids per-load EXEC save/restore branch sequences around the WMMAs.
        const int   n     = (M < 3) ? M : 0;
        const float bmask = (M < 3) ? 1.0f : 0.0f;
        v2f bfrag[3];
        #pragma unroll
        for (int j = 0; j < 3; ++j) {
            const int kb = j * 4;
            bfrag[j].x = fc_w[n * 12 + kb + koff]     * bmask;
            bfrag[j].y = fc_w[n * 12 + kb + koff + 1] * bmask;
        }

        #pragma unroll
        for (int g = 0; g < 2; ++g) {
            const int rowBase = waveId * 32 + g * 16;
            const float* ya = &s_y[(rowBase + M) * 13];
            v8f acc = {};
            #pragma unroll
            for (int j = 0; j < 3; ++j) {
                const int kb = j * 4;
                v2f av;
                av.x = ya[kb + koff];
                av.y = ya[kb + koff + 1];
                acc = __builtin_amdgcn_wmma_f32_16x16x4_f32(
                    /*neg_a=*/false, av, /*neg_b=*/false, bfrag[j],
                    /*c_mod=*/(short)0, acc, /*reuse_a=*/false, /*reuse_b=*/false);
            }
            // D layout: VGPR v holds M=v (lanes 0-15) / M=v+8 (lanes 16-31), N=lane&15.
            if (M < 3) {
                const int rOff = (lane < 16) ? 0 : 8;
                #pragma unroll
                for (int v = 0; v < 8; ++v)
                    s_logit[(rowBase + rOff + v) * 4 + M] = acc[v];
            }
        }
    }
    __syncthreads();

    // ---------------- Phase 5: + fc_b, softmax(3), store ----------------
    {
        const float l0 = s_logit[t * 4 + 0] + fc_b[0];
        const float l1 = s_logit[t * 4 + 1] + fc_b[1];
        const float l2 = s_logit[t * 4 + 2] + fc_b[2];
        const float mx = fmaxf(l0, fmaxf(l1, l2));
        const float e0 = expf(l0 - mx);
        const float e1 = expf(l1 - mx);
        const float e2 = expf(l2 - mx);
        const float inv = 1.0f / (e0 + e1 + e2);
        const long long r = blockRow0 + t;
        out[r * 3 + 0] = e0 * inv;
        out[r * 3 + 1] = e1 * inv;
        out[r * 3 + 2] = e2 * inv;
    }
}

extern "C" void kernel_launch(void* const* d_in, const int* in_sizes, int n_in,
                              void* d_out, int out_size, void* d_ws, size_t ws_size,
                              hipStream_t stream) {
    const float* x       = (const float*)d_in[0];
    const float* wh_rec  = (const float*)d_in[1];
    const float* wx_rec  = (const float*)d_in[2];
    const float* wx_eval = (const float*)d_in[3];
    const float* wh_eval = (const float*)d_in[4];
    const float* b_eval  = (const float*)d_in[5];
    const float* conv_w  = (const float*)d_in[6];
    const float* conv_b  = (const float*)d_in[7];
    const float* fc_w    = (const float*)d_in[8];
    const float* fc_b    = (const float*)d_in[9];
    float* out = (float*)d_out;

    const int rows = in_sizes[0] / 400;   // B = 131072
    const int grid = rows / RPB;          // 2048 blocks of 64 threads
    pde_fused_kernel<<<grid, RPB, 0, stream>>>(
        x, wh_rec, wx_rec, wx_eval, wh_eval, b_eval,
        conv_w, conv_b, fc_w, fc_b, out);
}